// layer_1_to_2_1580547974304
// MI455X (gfx1250) — compile-verified
//
#include <hip/hip_runtime.h>

// Problem constants (from reference): N=64, D=64, S=64, m=128
#define CN 64
#define CD 64
#define CS 64
#define CM 128

typedef float v2f __attribute__((ext_vector_type(2)));
typedef float v4f __attribute__((ext_vector_type(4)));
typedef float v8f __attribute__((ext_vector_type(8)));

// ---------------------------------------------------------------------------
// Kernel 1: sum_all[n,d] = sum_k inputs[n,d,k]   (one thread per (n,d) row)
// ---------------------------------------------------------------------------
__global__ void __launch_bounds__(256) sum_rows_kernel(const float* __restrict__ x,
                                                       float* __restrict__ sums) {
    const int idx = blockIdx.x * 256 + threadIdx.x;          // n*D + d
    const v4f* p = (const v4f*)(x + (size_t)idx * CM);
    float s = 0.f;
#pragma unroll
    for (int i = 0; i < CM / 4; ++i) {
        v4f v = p[i];
        s += v.x + v.y + v.z + v.w;
    }
    sums[idx] = s;
}

// ---------------------------------------------------------------------------
// Kernel 2: t2[n,s], t5[n,s] = (1/m) * sum_d coeffs[d,s,{1,4}] * sum_all[n,d]
// ---------------------------------------------------------------------------
__global__ void __launch_bounds__(256) t25_kernel(const float* __restrict__ coeffs,
                                                  const float* __restrict__ sums,
                                                  float* __restrict__ t2,
                                                  float* __restrict__ t5) {
    const int idx = blockIdx.x * 256 + threadIdx.x;          // n*S + s
    const int n = idx >> 6;
    const int s = idx & (CS - 1);
    float a2 = 0.f, a5 = 0.f;
    for (int d = 0; d < CD; ++d) {
        const float sa = sums[n * CD + d];
        const float* cp = coeffs + ((size_t)d * CS + s) * 5;
        a2 += cp[1] * sa;
        a5 += cp[4] * sa;
    }
    const float inv_m = 1.0f / (float)CM;
    t2[idx] = a2 * inv_m;
    t5[idx] = a5 * inv_m;
}

// ---------------------------------------------------------------------------
// Kernel 3: WMMA fp32 GEMM — t{1,3,4}[n,s,m] = sum_d coeffs[d,s,c] * x[n,d,m]
//
// One block per n (8 waves). Each wave computes 16x16 (S x m) tiles with
// V_WMMA_F32_16X16X4_F32, K-loop of 16 steps over D=64. Three accumulators
// (coeff slices 0,2,3) share the B operand (the inputs tile).
//
// VGPR layouts per CDNA5 ISA:
//   A (16x4): lane l -> M = l%16, K = 2*(l/16) + vgpr
//   B (4x16): lane l -> N = l%16, K = 2*(l/16) + vgpr
//   C/D (8 VGPRs): vgpr r, lane l -> M = r + 8*(l/16), N = l%16
// ---------------------------------------------------------------------------
__global__ void __launch_bounds__(256) gemm_wmma_kernel(const float* __restrict__ x,
                                                        const float* __restrict__ coeffs,
                                                        float* __restrict__ t1,
                                                        float* __restrict__ t3,
                                                        float* __restrict__ t4) {
    const int n    = blockIdx.x;
    const int wave = threadIdx.x >> 5;
    const int lane = threadIdx.x & 31;
    const int lo   = lane & 15;
    const int hi   = lane >> 4;

    for (int t = wave; t < 32; t += 8) {          // 4 S-tiles x 8 m-tiles
        const int s0 = (t >> 3) << 4;
        const int m0 = (t & 7) << 4;

        v8f acc1 = {};
        v8f acc3 = {};
        v8f acc4 = {};

        for (int k = 0; k < CD / 4; ++k) {
            const int d0 = 4 * k + 2 * hi;        // K for vgpr0 of this lane half
            // A: coeffs[d, s0+lo, c] for d = d0, d0+1
            const float* cp0 = coeffs + ((size_t)d0 * CS + (s0 + lo)) * 5;
            const float* cp1 = cp0 + (size_t)CS * 5;
            v2f a1 = { cp0[0], cp1[0] };
            v2f a3 = { cp0[2], cp1[2] };
            v2f a4 = { cp0[3], cp1[3] };
            // B: x[n, d, m0+lo] for d = d0, d0+1
            const float* xp = x + ((size_t)n * CD + d0) * CM + m0 + lo;
            v2f b = { xp[0], xp[CM] };

            acc1 = __builtin_amdgcn_wmma_f32_16x16x4_f32(false, a1, false, b,
                                                         (short)0, acc1, false, false);
            acc3 = __builtin_amdgcn_wmma_f32_16x16x4_f32(false, a3, false, b,
                                                         (short)0, acc3, false, false);
            acc4 = __builtin_amdgcn_wmma_f32_16x16x4_f32(false, a4, false, b,
                                                         (short)0, acc4, false, false);
        }

#pragma unroll
        for (int r = 0; r < 8; ++r) {
            const int s = s0 + r + 8 * hi;
            const size_t o = ((size_t)n * CS + s) * CM + m0 + lo;
            t1[o] = acc1[r];
            t3[o] = acc3[r];
            t4[o] = acc4[r];
        }
    }
}

// ---------------------------------------------------------------------------
// Kernel 4: assemble output (store-bandwidth bound, 256 MB)
//   out[n,s,i,j] = t3[n,s,j] + t4[n,s,i] + t5[n,s] + bias[s]
//                  + (i==j ? t1[n,s,i] + t2[n,s] : 0)
// One wave per row (n,s,i); each lane writes one 16B NT store.
// ---------------------------------------------------------------------------
__global__ void __launch_bounds__(256) assemble_kernel(const float* __restrict__ t1,
                                                       const float* __restrict__ t2,
                                                       const float* __restrict__ t3,
                                                       const float* __restrict__ t4,
                                                       const float* __restrict__ t5,
                                                       const float* __restrict__ bias,
                                                       float* __restrict__ out) {
    const int lane   = threadIdx.x & 31;
    const size_t row = (size_t)blockIdx.x * 8 + (threadIdx.x >> 5); // (n*S+s)*m + i
    const int i      = (int)(row & (CM - 1));
    const size_t ns  = row >> 7;                                    // n*S + s
    const int s      = (int)(ns & (CS - 1));

    const float base = t4[ns * CM + i] + t5[ns] + bias[s];

    v4f v = ((const v4f*)(t3 + ns * CM))[lane];
    v.x += base; v.y += base; v.z += base; v.w += base;

    if ((i >> 2) == lane) {                       // this lane's 4 j's contain j==i
        const float dg = t1[ns * CM + i] + t2[ns];
        const int c = i & 3;
        if      (c == 0) v.x += dg;
        else if (c == 1) v.y += dg;
        else if (c == 2) v.z += dg;
        else             v.w += dg;
    }

    __builtin_nontemporal_store(v, (v4f*)(out + row * CM) + lane);
}

// ---------------------------------------------------------------------------
extern "C" void kernel_launch(void* const* d_in, const int* in_sizes, int n_in,
                              void* d_out, int out_size, void* d_ws, size_t ws_size,
                              hipStream_t stream) {
    (void)in_sizes; (void)n_in; (void)out_size; (void)ws_size;

    const float* x      = (const float*)d_in[0];  // [N, D, m]
    const float* coeffs = (const float*)d_in[1];  // [D, S, 5]
    const float* bias   = (const float*)d_in[2];  // [1, S, 1, 1] -> [S]
    float* out = (float*)d_out;                   // [N, S, m, m]
    float* ws  = (float*)d_ws;

    const size_t NSM = (size_t)CN * CS * CM;      // 524288
    float* t1   = ws;
    float* t3   = ws + NSM;
    float* t4   = ws + 2 * NSM;
    float* t2   = ws + 3 * NSM;
    float* t5   = t2 + (size_t)CN * CS;
    float* sums = t5 + (size_t)CN * CS;           // total ~6.4 MB of workspace

    sum_rows_kernel<<<(CN * CD) / 256, 256, 0, stream>>>(x, sums);
    t25_kernel<<<(CN * CS) / 256, 256, 0, stream>>>(coeffs, sums, t2, t5);
    gemm_wmma_kernel<<<CN, 256, 0, stream>>>(x, coeffs, t1, t3, t4);
    assemble_kernel<<<(int)(NSM / 8), 256, 0, stream>>>(t1, t2, t3, t4, t5, bias, out);
}